// ddpOptimizer_41515153883557
// MI455X (gfx1250) — compile-verified
//
#include <hip/hip_runtime.h>

// IPDDP backward Riccati recursion on gfx1250 (MI455X).
// Single persistent workgroup (8 wave32s), sequential over T=8192 steps.
// FP32 WMMA 16x16x4 for all GEMMs, cooperative Gauss-Jordan for the
// 16x16 multi-RHS solve, TDM (tensor_load_to_lds) double-buffered staging.

typedef __attribute__((ext_vector_type(2))) float v2f;
typedef __attribute__((ext_vector_type(8))) float v8f;
typedef __attribute__((ext_vector_type(4))) unsigned int v4u;
typedef __attribute__((ext_vector_type(8))) int v8i;
typedef __attribute__((ext_vector_type(4))) int v4i;

#define T_HOR  8192
#define NSD    32
#define NTOT   48
#define NCONS  12
#define REGV   1.0f
#define SQ     49   // padded stride for row-indexed (frag_a) 48-col scratch
#define SQI    48   // natural stride for TDM-filled input buffers
#define SV     33   // padded stride for 32-col matrices

// ---- WMMA fragment helpers (wave32 layouts per CDNA5 ISA 7.12.2) ----
// A 16x4 f32: lanes 0-15 -> M=lane, K={0,1}; lanes 16-31 -> M=lane-16, K={2,3}
__device__ __forceinline__ v2f frag_a(const float* m, int stride, int r0, int k0) {
  int lane = threadIdx.x & 31;
  int r = r0 + (lane & 15);
  int k = k0 + ((lane >> 4) << 1);
  v2f a; a.x = m[r * stride + k]; a.y = m[r * stride + k + 1];
  return a;
}
// A-transposed load: A[m][k] = src[k][m]
__device__ __forceinline__ v2f frag_aT(const float* m, int stride, int r0, int k0) {
  int lane = threadIdx.x & 31;
  int r = r0 + (lane & 15);
  int k = k0 + ((lane >> 4) << 1);
  v2f a; a.x = m[k * stride + r]; a.y = m[(k + 1) * stride + r];
  return a;
}
// B 4x16 f32: lanes 0-15 -> N=lane, K={0,1}; lanes 16-31 -> N=lane-16, K={2,3}
__device__ __forceinline__ v2f frag_b(const float* m, int stride, int k0, int n0) {
  int lane = threadIdx.x & 31;
  int n = n0 + (lane & 15);
  int k = k0 + ((lane >> 4) << 1);
  v2f b; b.x = m[k * stride + n]; b.y = m[(k + 1) * stride + n];
  return b;
}
// C/D 16x16 f32: VGPR i -> M = i + (lane<16 ? 0 : 8), N = lane&15
__device__ __forceinline__ v8f frag_c(const float* m, int stride, int r0, int n0) {
  int lane = threadIdx.x & 31;
  int n = n0 + (lane & 15);
  int rb = r0 + ((lane >> 4) << 3);
  v8f c;
#pragma unroll
  for (int i = 0; i < 8; ++i) c[i] = m[(rb + i) * stride + n];
  return c;
}
__device__ __forceinline__ void store_c(float* m, int stride, int r0, int n0, v8f c) {
  int lane = threadIdx.x & 31;
  int n = n0 + (lane & 15);
  int rb = r0 + ((lane >> 4) << 3);
#pragma unroll
  for (int i = 0; i < 8; ++i) m[(rb + i) * stride + n] = c[i];
}
__device__ __forceinline__ v8f wmma4(v2f a, v2f b, v8f c) {
  // D = A(16x4) * B(4x16) + C, fp32 throughout.
  return __builtin_amdgcn_wmma_f32_16x16x4_f32(false, a, false, b, (short)0, c,
                                               false, false);
}

// ---- TDM: 1-D contiguous DMA of nelem f32 from global to LDS ----
// D# packing per CDNA5 ISA (groups 0/1; 2D form,
// groups 2 and 3 zero). Tracked by TENSORcnt.
__device__ __forceinline__ void tdm_copy_1d(const float* gsrc, float* ldst,
                                            int nelem) {
  unsigned long long ga = (unsigned long long)(uintptr_t)gsrc;
  unsigned int la = (unsigned int)(uintptr_t)ldst;  // flat low 32 = LDS offset
  v4u g0;
  g0[0] = 1u;                                       // count=1, user descriptor
  g0[1] = la;                                       // lds_addr
  g0[2] = (unsigned int)ga;                         // global_addr[31:0]
  g0[3] = (unsigned int)((ga >> 32) & 0x1FFFFFFu)   // global_addr[56:32]
          | (2u << 30);                             // type=2 ("image")
  v8i g1;
  g1[0] = (int)(2u << 16);                    // wg_mask=0, data_size=2 (4B)
  g1[1] = (int)(((unsigned)nelem & 0xFFFFu) << 16);  // tensor_dim0[15:0]
  g1[2] = (int)(1u << 16);                    // tensor_dim0[31:16]=0, tensor_dim1=1
  g1[3] = (int)((unsigned)nelem << 16);       // tensor_dim1 hi=0, tile_dim0=nelem
  g1[4] = 1;                                  // tile_dim1=1, tile_dim2=0
  g1[5] = nelem;                              // tensor_dim0_stride[31:0]
  g1[6] = 0;                                  // dim0_stride hi, dim1_stride lo
  g1[7] = 0;                                  // dim1_stride hi
  v4i gz4 = {0, 0, 0, 0};
  v8i gz8 = {0, 0, 0, 0, 0, 0, 0, 0};
  __builtin_amdgcn_tensor_load_to_lds(g0, g1, gz4, gz4, gz8, 0);
}

__global__ __launch_bounds__(256, 1)
void ipddp_bwd_kernel(const float* __restrict__ Q, const float* __restrict__ p,
                      const float* __restrict__ W, const float* __restrict__ F,
                      const float* __restrict__ s, const float* __restrict__ Qxx_term,
                      const float* __restrict__ qx_term,
                      float* __restrict__ o_ku, float* __restrict__ o_Ku,
                      float* __restrict__ o_dV, float* __restrict__ o_Vf,
                      float* __restrict__ o_vf) {
  // Carried state
  __shared__ float Vbuf[NSD * SV];       // V (32x32), symmetrized
  __shared__ float vvec[NSD];            // v (32)
  // Double-buffered per-step inputs (TDM-filled, natural stride 48)
  __shared__ float qbuf[2][NTOT * SQI];  // Q_t 48x48
  __shared__ float fbuf[2][NSD * SQI];   // F_t 32x48
  __shared__ float wbuf[2][NCONS * SQI]; // W_t 12x48
  __shared__ float pbuf[2][NTOT];        // p_t 48
  __shared__ float sbuf[2][16];          // s_t 12
  // Step scratch
  __shared__ float M1[NSD * SQ];         // V@F (32x48); reused as Vnew (32x32)
  __shared__ float QtS[NTOT * SQ];       // Qt (48x48)
  __shared__ float qtv[NTOT];            // qt (48)
  __shared__ float Aug[16 * SQ];         // [Quu+I | qu | Qxu^T]  (16x49)
  __shared__ float prow[52];             // scaled pivot row
  __shared__ float mcol[16];             // multiplier column
  __shared__ float kuS[16];              // ku
  __shared__ float KuS[16 * SV];         // Ku (16x32)
  __shared__ float dvacc[2];             // running sum of dV

  const int tid = threadIdx.x;
  const int w = tid >> 5;  // wave id (uniform per wave)

  // ---- init carried state ----
  for (int idx = tid; idx < NSD * NSD; idx += 256) {
    int i = idx >> 5, j = idx & 31;
    Vbuf[i * SV + j] = Qxx_term[idx];
  }
  if (tid < NSD) vvec[tid] = qx_term[tid];
  if (tid < 2) dvacc[tid] = 0.0f;

  // ---- preload step T-1 via TDM (wave 7 issues, waits, then barrier) ----
  if (w == 7) {
    const int t0 = T_HOR - 1;
    const int b = t0 & 1;
    tdm_copy_1d(Q + (size_t)t0 * 2304, qbuf[b], 2304);
    tdm_copy_1d(F + (size_t)t0 * 1536, fbuf[b], 1536);
    tdm_copy_1d(W + (size_t)t0 * 576, wbuf[b], 576);
    tdm_copy_1d(p + (size_t)t0 * 48, pbuf[b], 48);
    tdm_copy_1d(s + (size_t)t0 * 12, sbuf[b], 12);
    __builtin_amdgcn_s_wait_tensorcnt(0);
  }
  __syncthreads();

  for (int t = T_HOR - 1; t >= 0; --t) {
    const int cur = t & 1;
    const int nxt = cur ^ 1;
    const float* qb = qbuf[cur];
    const float* fb = fbuf[cur];
    const float* wb = wbuf[cur];
    const float* pb = pbuf[cur];
    const float* sb = sbuf[cur];

    // -- wave 7: kick off DMA of step t-1 into the idle buffer half --
    if (t > 0 && w == 7) {
      tdm_copy_1d(Q + (size_t)(t - 1) * 2304, qbuf[nxt], 2304);
      tdm_copy_1d(F + (size_t)(t - 1) * 1536, fbuf[nxt], 1536);
      tdm_copy_1d(W + (size_t)(t - 1) * 576, wbuf[nxt], 576);
      tdm_copy_1d(p + (size_t)(t - 1) * 48, pbuf[nxt], 48);
      tdm_copy_1d(s + (size_t)(t - 1) * 12, sbuf[nxt], 12);
    }
    // -- L2 prefetch 5 steps ahead (one 128B line per thread) --
    if (t > 4) {
      const int tp = t - 5;
      if (tid < 72)       __builtin_prefetch(Q + (size_t)tp * 2304 + tid * 32);
      else if (tid < 120) __builtin_prefetch(F + (size_t)tp * 1536 + (tid - 72) * 32);
      else if (tid < 138) __builtin_prefetch(W + (size_t)tp * 576 + (tid - 120) * 32);
      else if (tid < 140) __builtin_prefetch(p + (size_t)tp * 48 + (tid - 138) * 32);
    }

    // ---- Phase A: waves 0-5: M1 = V @ F (32x48, 6 tiles);
    //               waves 6-7: qt = p + F^T v + W^T s (48) ----
    if (w < 6) {
      int tm = w / 3, tn = w - tm * 3;
      v8f acc = {};
#pragma unroll
      for (int kk = 0; kk < 8; ++kk) {
        v2f a = frag_a(Vbuf, SV, tm * 16, kk * 4);
        v2f b = frag_b(fb, SQI, kk * 4, tn * 16);
        acc = wmma4(a, b, acc);
      }
      store_c(M1, SQ, tm * 16, tn * 16, acc);
    } else {
      int i = tid - 192;
      if (i < NTOT) {
        float acc = pb[i];
#pragma unroll 4
        for (int k = 0; k < NSD; ++k) acc += fb[k * SQI + i] * vvec[k];
#pragma unroll
        for (int j = 0; j < NCONS; ++j) acc += wb[j * SQI + i] * sb[j];
        qtv[i] = acc;
      }
    }
    __syncthreads();

    // ---- Phase B: Qt = Qt0 + F^T @ M1 (48x48, 9 tiles over 8 waves) ----
    for (int tile = w; tile < 9; tile += 8) {
      int tm = tile / 3, tn = tile - tm * 3;
      v8f acc = frag_c(qb, SQI, tm * 16, tn * 16);  // C initialized with Qt0 tile
#pragma unroll
      for (int kk = 0; kk < 8; ++kk) {
        v2f a = frag_aT(fb, SQI, tm * 16, kk * 4);  // (F^T)[m][k] = F[k][m]
        v2f b = frag_b(M1, SQ, kk * 4, tn * 16);
        acc = wmma4(a, b, acc);
      }
      store_c(QtS, SQ, tm * 16, tn * 16, acc);
    }
    __syncthreads();

    // ---- Phase C: build Aug = [Quu+REG*I | qu | Qxu^T], Gauss-Jordan ----
    for (int idx = tid; idx < 16 * SQ; idx += 256) {
      int r = idx / SQ, c = idx - r * SQ;
      float val;
      if (c < 16)
        val = QtS[(32 + r) * SQ + (32 + c)] + (r == c ? REGV : 0.0f);
      else if (c == 16)
        val = qtv[32 + r];
      else
        val = QtS[(c - 17) * SQ + (32 + r)];  // Qxu^T[r][c-17] = Qt[c-17][32+r]
      Aug[idx] = val;
    }
    __syncthreads();
    for (int pp = 0; pp < 16; ++pp) {
      if (tid < SQ) {
        float inv = 1.0f / Aug[pp * SQ + pp];
        prow[tid] = Aug[pp * SQ + tid] * inv;
      } else if (tid >= 64 && tid < 80) {
        mcol[tid - 64] = Aug[(tid - 64) * SQ + pp];
      }
      __syncthreads();
      for (int idx = tid; idx < 16 * SQ; idx += 256) {
        int r = idx / SQ, c = idx - r * SQ;
        Aug[idx] = (r == pp) ? prow[c] : (Aug[idx] - mcol[r] * prow[c]);
      }
      __syncthreads();
    }
    // sol = Aug[:,16:49]; ku = -sol[:,0]; Ku = -sol[:,1:]
    if (tid < 16) {
      float kv = -Aug[tid * SQ + 16];
      kuS[tid] = kv;
      o_ku[t * 16 + tid] = kv;
    }
    for (int idx = tid; idx < 16 * 32; idx += 256) {
      int r = idx >> 5, j = idx & 31;
      float kv = -Aug[r * SQ + 17 + j];
      KuS[r * SV + j] = kv;
      o_Ku[t * 512 + idx] = kv;
    }
    __syncthreads();

    // ---- Phase D: waves 0-3: Vnew = Qxx + Qxu@Ku (32x32, 4 tiles);
    //               wave 4: vn = qx + Qxu@ku; wave 5: dV dots ----
    if (w < 4) {
      int tm = w >> 1, tn = w & 1;
      v8f acc = frag_c(QtS, SQ, tm * 16, tn * 16);  // C = Qxx tile
#pragma unroll
      for (int kk = 0; kk < 4; ++kk) {
        v2f a = frag_a(QtS, SQ, tm * 16, 32 + kk * 4);  // Qxu
        v2f b = frag_b(KuS, SV, kk * 4, tn * 16);
        acc = wmma4(a, b, acc);
      }
      store_c(M1, SQ, tm * 16, tn * 16, acc);  // M1 reused as Vnew
    } else if (w == 4) {
      int i = tid - 128;  // 0..31
      float acc = qtv[i];
#pragma unroll
      for (int k = 0; k < 16; ++k) acc += QtS[i * SQ + 32 + k] * kuS[k];
      vvec[i] = acc;  // v for next (earlier-time) step
    } else if (w == 5) {
      int lane = tid & 31;
      int k = lane & 15;
      float tmp = REGV * kuS[k];
#pragma unroll
      for (int j = 0; j < 16; ++j) tmp += QtS[(32 + k) * SQ + 32 + j] * kuS[j];
      float p0 = kuS[k] * qtv[32 + k];       // ku . qu
      float p1 = 0.5f * kuS[k] * tmp;        // 0.5 ku . (Quu ku)
      if (lane >= 16) { p0 = 0.0f; p1 = 0.0f; }
#pragma unroll
      for (int off = 8; off >= 1; off >>= 1) {
        p0 += __shfl_xor(p0, off, 32);
        p1 += __shfl_xor(p1, off, 32);
      }
      if (lane == 0) { dvacc[0] += p0; dvacc[1] += p1; }
    }
    __syncthreads();

    // ---- Phase E: symmetrize into carried V ----
    for (int idx = tid; idx < NSD * NSD; idx += 256) {
      int i = idx >> 5, j = idx & 31;
      Vbuf[i * SV + j] = 0.5f * (M1[i * SQ + j] + M1[j * SQ + i]);
    }

    // ---- wave 7: ensure next-step DMA has landed, then iteration barrier ----
    if (t > 0 && w == 7) __builtin_amdgcn_s_wait_tensorcnt(0);
    __syncthreads();
  }

  // ---- epilogue: Vf, vf, summed dV ----
  for (int idx = tid; idx < NSD * NSD; idx += 256) {
    int i = idx >> 5, j = idx & 31;
    o_Vf[idx] = Vbuf[i * SV + j];
  }
  if (tid < NSD) o_vf[tid] = vvec[tid];
  if (tid < 2) o_dV[tid] = dvacc[tid];
}

extern "C" void kernel_launch(void* const* d_in, const int* in_sizes, int n_in,
                              void* d_out, int out_size, void* d_ws, size_t ws_size,
                              hipStream_t stream) {
  (void)in_sizes; (void)n_in; (void)d_ws; (void)ws_size; (void)out_size;
  // setup_inputs order: Q, p, W, F, c(unused), s, Qxx_term, qx_term
  const float* Q = (const float*)d_in[0];
  const float* p = (const float*)d_in[1];
  const float* W = (const float*)d_in[2];
  const float* F = (const float*)d_in[3];
  const float* s = (const float*)d_in[5];
  const float* Qxx = (const float*)d_in[6];
  const float* qx = (const float*)d_in[7];

  float* out = (float*)d_out;
  float* o_ku = out;                              // 8192*16
  float* o_Ku = out + 8192 * 16;                  // 8192*512
  float* o_dV = out + 8192 * 16 + 8192 * 512;     // 2
  float* o_Vf = o_dV + 2;                         // 1024
  float* o_vf = o_Vf + 1024;                      // 32

  hipLaunchKernelGGL(ipddp_bwd_kernel, dim3(1), dim3(256), 0, stream,
                     Q, p, W, F, s, Qxx, qx, o_ku, o_Ku, o_dV, o_Vf, o_vf);
}